// GINENet_69045894250554
// MI455X (gfx1250) — compile-verified
//
#include <hip/hip_runtime.h>
#include <hip/hip_bf16.h>

// ---------------------------------------------------------------------------
// GINE GNN forward for MI455X (gfx1250, wave32).
// Dense node-feature GEMMs use V_WMMA_F32_16X16X4_F32 (fp32 precision, WMMA
// path). Edge gather/scatter is the bandwidth-bound part and uses float4
// loads + global float atomics.
// ---------------------------------------------------------------------------

#define NNODES 50000
#define NEDGES 800000
#define BN_EPS 1e-5f

typedef __attribute__((ext_vector_type(2))) float v2f;
typedef __attribute__((ext_vector_type(8))) float v8f;

// ---------------------------------------------------------------------------
// Edge kernel: msg = relu(x[src] + ea*lin_w + lin_b); atomic scatter-add to
// agg[dst]. 24 float4-chunks cover the 96 feature dims; consecutive threads
// handle consecutive chunks of one edge (coalesced row reads).
// ---------------------------------------------------------------------------
__global__ __launch_bounds__(256)
void edge_scatter_kernel(const float* __restrict__ x, int ldx,
                         const float* __restrict__ ea,
                         const int* __restrict__ src,
                         const int* __restrict__ dst,
                         const float* __restrict__ lw,
                         const float* __restrict__ lb,
                         float* __restrict__ agg)
{
    const int CHUNKS = 24;  // 96 / 4
    int gid = blockIdx.x * blockDim.x + threadIdx.x;
    if (gid >= NEDGES * CHUNKS) return;
    int j = gid / CHUNKS;
    int d = (gid % CHUNKS) * 4;

    int   s = src[j];
    int   t = dst[j];
    float a = ea[j];

    float4 xv  = *reinterpret_cast<const float4*>(x + (size_t)s * ldx + d);
    float4 lw4 = *reinterpret_cast<const float4*>(lw + d);
    float4 lb4 = *reinterpret_cast<const float4*>(lb + d);

    float m0 = fmaxf(xv.x + a * lw4.x + lb4.x, 0.0f);
    float m1 = fmaxf(xv.y + a * lw4.y + lb4.y, 0.0f);
    float m2 = fmaxf(xv.z + a * lw4.z + lb4.z, 0.0f);
    float m3 = fmaxf(xv.w + a * lw4.w + lb4.w, 0.0f);

    float* ap = agg + (size_t)t * 96 + d;
    atomicAdd(ap + 0, m0);
    atomicAdd(ap + 1, m1);
    atomicAdd(ap + 2, m2);
    atomicAdd(ap + 3, m3);
}

// ---------------------------------------------------------------------------
// WMMA fp32 GEMM: out[i,n] = relu( sum_k (A0[i,k] (+A1[i,k])) * W[k,n] + b[n] )
// One wave32 per 16x16 output tile. K stepped by 4 via V_WMMA_F32_16X16X4_F32.
// ISA layouts (CDNA5 ISA):
//   A 16x4: lanes 0-15 -> row=lane,   {K=k0,k0+1}; lanes 16-31 -> {k0+2,k0+3}
//   B 4x16: lanes 0-15 -> col=lane,   {K=k0,k0+1}; lanes 16-31 -> {k0+2,k0+3}
//   C/D:    VGPR r -> row = r + 8*hi, col = lane&15
// EXEC is all-ones (no divergence before the WMMA), as required.
// ---------------------------------------------------------------------------
template <int K, bool HASADD>
__global__ __launch_bounds__(32)
void gemm_wmma_f32_kernel(const float* __restrict__ A0, int lda0,
                          const float* __restrict__ A1, int lda1,
                          const float* __restrict__ W,  int ldw,
                          const float* __restrict__ bias,
                          float* __restrict__ out, int ldo)
{
    int lane = threadIdx.x;         // 0..31
    int lo   = lane & 15;
    int hi   = lane >> 4;
    int row0 = blockIdx.x * 16;
    int col0 = blockIdx.y * 16;

    const float* arow0 = A0 + (size_t)(row0 + lo) * lda0;
    const float* arow1 = HASADD ? (A1 + (size_t)(row0 + lo) * lda1) : nullptr;

    v8f c = {};
#pragma unroll
    for (int k0 = 0; k0 < K; k0 += 4) {
        int ka = k0 + 2 * hi;
        v2f a, b;
        a.x = arow0[ka];
        a.y = arow0[ka + 1];
        if (HASADD) {
            a.x += arow1[ka];
            a.y += arow1[ka + 1];
        }
        b.x = W[(size_t)ka * ldw + col0 + lo];
        b.y = W[(size_t)(ka + 1) * ldw + col0 + lo];
        c = __builtin_amdgcn_wmma_f32_16x16x4_f32(
                /*neg_a=*/false, a, /*neg_b=*/false, b,
                /*c_mod=*/(short)0, c, /*reuse_a=*/false, /*reuse_b=*/false);
    }

    float bv = bias[col0 + lo];
#pragma unroll
    for (int r = 0; r < 8; ++r) {
        int row = row0 + r + 8 * hi;
        out[(size_t)row * ldo + col0 + lo] = fmaxf(c[r] + bv, 0.0f);
    }
}

// ---------------------------------------------------------------------------
// BatchNorm batch statistics: per-column sum and sum of squares.
// blockDim.x == C (96/64/16); blocks stride over rows; one atomicAdd per
// (block, column).
// ---------------------------------------------------------------------------
__global__ void col_stats_kernel(const float* __restrict__ in, int ld, int n,
                                 float* __restrict__ sum,
                                 float* __restrict__ sumsq)
{
    int c = threadIdx.x;
    float s = 0.0f, s2 = 0.0f;
    for (int row = blockIdx.x; row < n; row += gridDim.x) {
        float v = in[(size_t)row * ld + c];
        s  += v;
        s2 += v * v;
    }
    atomicAdd(&sum[c],   s);
    atomicAdd(&sumsq[c], s2);
}

__global__ void bn_finalize_kernel(const float* __restrict__ sum,
                                   const float* __restrict__ sumsq,
                                   const float* __restrict__ g,
                                   const float* __restrict__ beta,
                                   float* __restrict__ scale,
                                   float* __restrict__ shift, float invN)
{
    int c = threadIdx.x;
    float mu  = sum[c] * invN;
    float var = sumsq[c] * invN - mu * mu;   // biased variance (matches ref)
    float inv = rsqrtf(var + BN_EPS);
    float sc  = g[c] * inv;
    scale[c] = sc;
    shift[c] = beta[c] - mu * sc;
}

__global__ __launch_bounds__(256)
void bn_apply_kernel(const float* __restrict__ in, int ldi,
                     float* __restrict__ out, int ldo,
                     const float* __restrict__ scale,
                     const float* __restrict__ shift, int n, int C)
{
    int gid = blockIdx.x * blockDim.x + threadIdx.x;
    if (gid >= n * C) return;
    int row = gid / C;
    int c   = gid % C;
    out[(size_t)row * ldo + c] = in[(size_t)row * ldi + c] * scale[c] + shift[c];
}

// ---------------------------------------------------------------------------
// Host orchestration
// ---------------------------------------------------------------------------
extern "C" void kernel_launch(void* const* d_in, const int* in_sizes, int n_in,
                              void* d_out, int out_size, void* d_ws, size_t ws_size,
                              hipStream_t stream)
{
    (void)in_sizes; (void)n_in; (void)out_size; (void)ws_size;

    const float* x          = (const float*)d_in[0];   // [N,96]
    const float* edge_attr  = (const float*)d_in[1];   // [E]
    const int*   edge_index = (const int*)  d_in[2];   // [2,E]
    const float* lin_w      = (const float*)d_in[3];   // [96]
    const float* lin_b      = (const float*)d_in[4];   // [96]
    const float* c1_w  = (const float*)d_in[5];
    const float* c1_b  = (const float*)d_in[6];
    const float* c1_g  = (const float*)d_in[7];
    const float* c1_be = (const float*)d_in[8];
    const float* c2_w  = (const float*)d_in[9];
    const float* c2_b  = (const float*)d_in[10];
    const float* c2_g  = (const float*)d_in[11];
    const float* c2_be = (const float*)d_in[12];
    const float* l1_w  = (const float*)d_in[13];
    const float* l1_b  = (const float*)d_in[14];
    const float* l1_g  = (const float*)d_in[15];
    const float* l1_be = (const float*)d_in[16];
    const float* ma_w  = (const float*)d_in[17];
    const float* ma_b  = (const float*)d_in[18];
    const float* ma_g  = (const float*)d_in[19];
    const float* ma_be = (const float*)d_in[20];
    const float* mb_w  = (const float*)d_in[21];
    const float* mb_b  = (const float*)d_in[22];
    const float* mb_g  = (const float*)d_in[23];
    const float* mb_be = (const float*)d_in[24];

    const int* src = edge_index;
    const int* dst = edge_index + NEDGES;

    const size_t N = NNODES;

    // Workspace layout (floats): ~90 MB total
    float* ws    = (float*)d_ws;
    float* agg   = ws;                 // [N,96] scatter accumulator
    float* bufP  = agg  + N * 96;      // [N,96] ping
    float* bufQ  = bufP + N * 96;      // [N,96] pong
    float* xcat  = bufQ + N * 96;      // [N,160] concat(x1, x2)
    float* stats = xcat + N * 160;     // 5 layers x {sum,sq,scale,shift}[96]

    const float invN = 1.0f / (float)N;
    const int rowTiles = NNODES / 16;  // 3125, exact

    auto statsL = [&](int L) { return stats + (size_t)L * 384; };

    // Zero accumulators (stream-ordered; legal inside graph capture)
    hipMemsetAsync(agg,   0, N * 96 * sizeof(float), stream);
    hipMemsetAsync(stats, 0, 5 * 384 * sizeof(float), stream);

    const int edgeThreads = NEDGES * 24;
    const int edgeBlocks  = (edgeThreads + 255) / 256;

    // ---- conv1: agg = segsum(relu(x[src]+e)) ; h = relu((x+agg)W1+b1); BN
    edge_scatter_kernel<<<edgeBlocks, 256, 0, stream>>>(
        x, 96, edge_attr, src, dst, lin_w, lin_b, agg);

    gemm_wmma_f32_kernel<96, true><<<dim3(rowTiles, 6), 32, 0, stream>>>(
        x, 96, agg, 96, c1_w, 96, c1_b, bufP, 96);

    col_stats_kernel<<<256, 96, 0, stream>>>(bufP, 96, NNODES,
                                             statsL(0), statsL(0) + 96);
    bn_finalize_kernel<<<1, 96, 0, stream>>>(statsL(0), statsL(0) + 96,
                                             c1_g, c1_be,
                                             statsL(0) + 192, statsL(0) + 288, invN);
    // x1 -> xcat[:, 0:96]
    bn_apply_kernel<<<(NNODES * 96 + 255) / 256, 256, 0, stream>>>(
        bufP, 96, xcat, 160, statsL(0) + 192, statsL(0) + 288, NNODES, 96);

    // ---- conv2: agg = segsum(relu(x1[src]+e)) ; h = relu((x1+agg)W2+b2); BN
    hipMemsetAsync(agg, 0, N * 96 * sizeof(float), stream);
    edge_scatter_kernel<<<edgeBlocks, 256, 0, stream>>>(
        xcat, 160, edge_attr, src, dst, lin_w, lin_b, agg);

    gemm_wmma_f32_kernel<96, true><<<dim3(rowTiles, 4), 32, 0, stream>>>(
        xcat, 160, agg, 96, c2_w, 64, c2_b, bufP, 64);

    col_stats_kernel<<<256, 64, 0, stream>>>(bufP, 64, NNODES,
                                             statsL(1), statsL(1) + 96);
    bn_finalize_kernel<<<1, 64, 0, stream>>>(statsL(1), statsL(1) + 96,
                                             c2_g, c2_be,
                                             statsL(1) + 192, statsL(1) + 288, invN);
    // x2 -> xcat[:, 96:160]
    bn_apply_kernel<<<(NNODES * 64 + 255) / 256, 256, 0, stream>>>(
        bufP, 64, xcat + 96, 160, statsL(1) + 192, statsL(1) + 288, NNODES, 64);

    // ---- lin1: h = relu(concat(x1,x2) @ W + b); BN (in place)
    gemm_wmma_f32_kernel<160, false><<<dim3(rowTiles, 6), 32, 0, stream>>>(
        xcat, 160, nullptr, 0, l1_w, 96, l1_b, bufQ, 96);

    col_stats_kernel<<<256, 96, 0, stream>>>(bufQ, 96, NNODES,
                                             statsL(2), statsL(2) + 96);
    bn_finalize_kernel<<<1, 96, 0, stream>>>(statsL(2), statsL(2) + 96,
                                             l1_g, l1_be,
                                             statsL(2) + 192, statsL(2) + 288, invN);
    bn_apply_kernel<<<(NNODES * 96 + 255) / 256, 256, 0, stream>>>(
        bufQ, 96, bufQ, 96, statsL(2) + 192, statsL(2) + 288, NNODES, 96);

    // ---- mlp1a: relu(h @ W + b); BN (in place)
    gemm_wmma_f32_kernel<96, false><<<dim3(rowTiles, 6), 32, 0, stream>>>(
        bufQ, 96, nullptr, 0, ma_w, 96, ma_b, bufP, 96);

    col_stats_kernel<<<256, 96, 0, stream>>>(bufP, 96, NNODES,
                                             statsL(3), statsL(3) + 96);
    bn_finalize_kernel<<<1, 96, 0, stream>>>(statsL(3), statsL(3) + 96,
                                             ma_g, ma_be,
                                             statsL(3) + 192, statsL(3) + 288, invN);
    bn_apply_kernel<<<(NNODES * 96 + 255) / 256, 256, 0, stream>>>(
        bufP, 96, bufP, 96, statsL(3) + 192, statsL(3) + 288, NNODES, 96);

    // ---- mlp1b: relu(h @ W + b); BN -> d_out [N,16]
    gemm_wmma_f32_kernel<96, false><<<dim3(rowTiles, 1), 32, 0, stream>>>(
        bufP, 96, nullptr, 0, mb_w, 16, mb_b, bufQ, 16);

    col_stats_kernel<<<256, 16, 0, stream>>>(bufQ, 16, NNODES,
                                             statsL(4), statsL(4) + 96);
    bn_finalize_kernel<<<1, 16, 0, stream>>>(statsL(4), statsL(4) + 96,
                                             mb_g, mb_be,
                                             statsL(4) + 192, statsL(4) + 288, invN);
    bn_apply_kernel<<<(NNODES * 16 + 255) / 256, 256, 0, stream>>>(
        bufQ, 16, (float*)d_out, 16, statsL(4) + 192, statsL(4) + 288, NNODES, 16);
}